// TokenMixer_15788299780170
// MI455X (gfx1250) — compile-verified
//
#include <hip/hip_runtime.h>
#include <stdint.h>

#define LDS_AS __attribute__((address_space(3)))

constexpr int BUF  = 512;   // slots per class
constexpr int DIM  = 256;   // token dim
constexpr int KTOP = 16;    // k

// ---------------------------------------------------------------------------
// Kernel A: per-class top-16 of areas[c][0..511] -> topi[c][0..15]
// One block per class. Iterative extract-max, stable (smaller index wins ties)
// to match jax.lax.top_k ordering. 1000 blocks, trivial cost (~2 MB read).
// ---------------------------------------------------------------------------
__global__ void __launch_bounds__(256)
topk16_kernel(const float* __restrict__ areas, int* __restrict__ topi) {
  __shared__ float vals[BUF];
  __shared__ float rv[256];
  __shared__ int   ri[256];

  const int c = blockIdx.x;
  const int t = threadIdx.x;

  vals[t]       = areas[(size_t)c * BUF + t];
  vals[t + 256] = areas[(size_t)c * BUF + t + 256];
  __syncthreads();

  for (int iter = 0; iter < KTOP; ++iter) {
    float v0 = vals[t];
    float v1 = vals[t + 256];
    float bv = v0; int bi = t;
    if (v1 > v0) { bv = v1; bi = t + 256; }   // tie -> keep smaller index
    rv[t] = bv; ri[t] = bi;
    __syncthreads();
#pragma unroll
    for (int s = 128; s > 0; s >>= 1) {
      if (t < s) {
        float ov = rv[t + s]; int oi = ri[t + s];
        if (ov > rv[t] || (ov == rv[t] && oi < ri[t])) { rv[t] = ov; ri[t] = oi; }
      }
      __syncthreads();
    }
    if (t == 0) {
      topi[c * KTOP + iter] = ri[0];
      vals[ri[0]] = -__builtin_inff();        // remove winner
    }
    __syncthreads();
  }
}

// ---------------------------------------------------------------------------
// Kernel B: gather. One block per query (16 rows x 256 f32 = 16 KB out).
// CDNA5 async DMA path: global -> LDS (GLOBAL_LOAD_ASYNC_TO_LDS_B128),
// s_wait_asynccnt, LDS -> global (GLOBAL_STORE_ASYNC_FROM_LDS_B128).
// Each lane owns disjoint 16B LDS slots, so per-wave ASYNCcnt ordering is
// sufficient: no workgroup barriers between the load and store phases.
//
// Cache policy (the key roofline lever):
//   loads  th:TH_LOAD_HT  -> keep the ~16 MB top-row working set resident in
//                            L2 (each class's rows are re-read ~65x),
//   stores th:TH_STORE_NT -> the 1 GiB output is write-once; don't let it
//                            evict the hot read set from L2.
// ---------------------------------------------------------------------------
__global__ void __launch_bounds__(256)
gather16_kernel(const float* __restrict__ buffer,
                const int* __restrict__ labels,
                const int* __restrict__ topi,
                float* __restrict__ out) {
  __shared__ int   rows[KTOP];
  __shared__ float stage[KTOP * DIM];   // 16 KB staging

  const int q = blockIdx.x;
  const int t = threadIdx.x;
  const int label = labels[q];

  if (t < KTOP) rows[t] = topi[label * KTOP + t];
  __syncthreads();

  const float* src_base = buffer + (size_t)label * (BUF * DIM);
  float*       dst_base = out    + (size_t)q     * (KTOP * DIM);

  // Wave-relative LDS byte address of the staging buffer (flat->group cast:
  // low 32 bits are the group offset, ISA sec 10.2).
  const unsigned lds_base = (unsigned)(uintptr_t)(LDS_AS float*)stage;

  // Phase 1: async global -> LDS, 4 x b128 per lane (ASYNCcnt <= 4 per wave)
#pragma unroll
  for (int i = 0; i < 4; ++i) {
    const int idx = i * 256 + t;            // float4 slot 0..1023
    const int row = idx >> 6;
    const int c4  = idx & 63;
    const float* src = src_base + (size_t)rows[row] * DIM + c4 * 4;
    const unsigned lds_off = lds_base + (unsigned)idx * 16u;
    asm volatile("global_load_async_to_lds_b128 %0, %1, off th:TH_LOAD_HT"
                 :: "v"(lds_off),
                    "v"((unsigned long long)(uintptr_t)src)
                 : "memory");
  }

  asm volatile("s_wait_asynccnt 0x0" ::: "memory");

  // Phase 2: async LDS -> global, non-temporal (write-once stream)
#pragma unroll
  for (int i = 0; i < 4; ++i) {
    const int idx = i * 256 + t;
    const int row = idx >> 6;
    const int c4  = idx & 63;
    float* dst = dst_base + row * DIM + c4 * 4;
    const unsigned lds_off = lds_base + (unsigned)idx * 16u;
    asm volatile("global_store_async_from_lds_b128 %0, %1, off th:TH_STORE_NT"
                 :: "v"((unsigned long long)(uintptr_t)dst),
                    "v"(lds_off)
                 : "memory");
  }

  asm volatile("s_wait_asynccnt 0x0" ::: "memory");
}

// ---------------------------------------------------------------------------
// Inputs (setup_inputs order):
//   d_in[0] buffer  [1000, 512, 256] f32
//   d_in[1] areas   [1000, 512]      f32
//   d_in[2] pointer [1000]           f32   (validity check only; unused)
//   d_in[3] obj_labels [65536]       int
//   d_in[4] k       [1]              int   (== 16, baked into kernels)
// d_out: [N, 16, 256] f32.  d_ws: top-k index table (1000*16 ints = 64 KB).
// ---------------------------------------------------------------------------
extern "C" void kernel_launch(void* const* d_in, const int* in_sizes, int n_in,
                              void* d_out, int out_size, void* d_ws, size_t ws_size,
                              hipStream_t stream) {
  (void)n_in; (void)out_size; (void)ws_size;

  const float* buffer = (const float*)d_in[0];
  const float* areas  = (const float*)d_in[1];
  const int*   labels = (const int*)d_in[3];

  const int C = in_sizes[2];   // number of classes (pointer has C elements)
  const int N = in_sizes[3];   // number of queries

  int* topi = (int*)d_ws;      // [C, 16]

  topk16_kernel<<<dim3(C), dim3(256), 0, stream>>>(areas, topi);
  gather16_kernel<<<dim3(N), dim3(256), 0, stream>>>(buffer, labels, topi,
                                                     (float*)d_out);
}